// GCNNet_53171695124573
// MI455X (gfx1250) — compile-verified
//
#include <hip/hip_runtime.h>
#include <hip/hip_bf16.h>

typedef __attribute__((ext_vector_type(2))) float v2f;
typedef __attribute__((ext_vector_type(8))) float v8f;

#define SLOPE 0.02f
#define NF 128

// ---------------- normalization precompute ----------------

__global__ void deg_init_kernel(float* deg, int n) {
    int i = blockIdx.x * blockDim.x + threadIdx.x;
    if (i < n) deg[i] = 1.0f;  // self-loop contributes 1
}

__global__ void deg_count_kernel(const long long* __restrict__ dst, float* deg, int E) {
    int e = blockIdx.x * blockDim.x + threadIdx.x;
    if (e < E) atomicAdd(&deg[(int)dst[e]], 1.0f);
}

__global__ void dinv_kernel(float* deg, int n) {
    int i = blockIdx.x * blockDim.x + threadIdx.x;
    if (i < n) deg[i] = rsqrtf(deg[i]);
}

__global__ void edge_norm_kernel(const long long* __restrict__ src,
                                 const long long* __restrict__ dst,
                                 const float* __restrict__ dinv,
                                 float* __restrict__ norm, int E) {
    int e = blockIdx.x * blockDim.x + threadIdx.x;
    if (e < E) norm[e] = dinv[(int)src[e]] * dinv[(int)dst[e]];
}

// ---------------- dense GEMM via fp32 WMMA ----------------
// T[M,128] = (lrelu?)(Hin[M,K]) @ W[K,128]
// Persistent waves: each wave owns one 16-wide N tile, preloads its K x 16
// B panel into registers (K/4 v2f = up to 64 VGPRs), then streams M tiles.
// Steady-state inner loop: 1 global_load_b64 (A pair) + 1 v_wmma per K-step.
template <int K, bool LRELU>
__global__ __launch_bounds__(256) void gemm_wmma_kernel(
    const float* __restrict__ Hin, const float* __restrict__ W,
    float* __restrict__ T, int Mtiles) {
    const int lane = threadIdx.x & 31;
    const int hi   = lane >> 4;               // selects K pair {2hi, 2hi+1} within step
    const int lo   = lane & 15;
    const int bcol = (threadIdx.x >> 5) * 16 + lo;  // wave id 0..7 -> N tile

    constexpr int KS = K / 4;
    v2f breg[KS];
#pragma unroll
    for (int ks = 0; ks < KS; ++ks) {
        const int ka = 4 * ks + 2 * hi;
        breg[ks].x = W[(size_t)ka * NF + bcol];
        breg[ks].y = W[(size_t)(ka + 1) * NF + bcol];
    }

    for (int mt = blockIdx.x; mt < Mtiles; mt += gridDim.x) {
        const float* __restrict__ arow_p = Hin + (size_t)(mt * 16 + lo) * K;
        v8f acc = {};
#pragma unroll
        for (int ks = 0; ks < KS; ++ks) {
            const int ka = 4 * ks + 2 * hi;
            float a0 = arow_p[ka];
            float a1 = arow_p[ka + 1];
            if (LRELU) {
                a0 = fmaxf(a0, 0.0f) + SLOPE * fminf(a0, 0.0f);
                a1 = fmaxf(a1, 0.0f) + SLOPE * fminf(a1, 0.0f);
            }
            v2f a; a.x = a0; a.y = a1;
            acc = __builtin_amdgcn_wmma_f32_16x16x4_f32(
                /*neg_a=*/false, a, /*neg_b=*/false, breg[ks],
                /*c_mod=*/(short)0, acc, /*reuse_a=*/false, /*reuse_b=*/false);
        }
        // D layout: VGPR i -> row M = i + 8*hi, col N = lo
        float* outp = T + (size_t)(mt * 16) * NF + bcol;
#pragma unroll
        for (int i = 0; i < 8; ++i) {
            outp[(size_t)(i + 8 * hi) * NF] = acc[i];
        }
    }
}

// ---------------- aggregation ----------------
// H[i] = bias + T[i] * dinv[i]^2   (self-loop message + bias; non-atomic init)
__global__ void agg_init_kernel(const float* __restrict__ T,
                                const float* __restrict__ dinv,
                                const float* __restrict__ bias,
                                float* __restrict__ H, int n) {
    int idx = blockIdx.x * blockDim.x + threadIdx.x;
    int i = idx >> 5;                 // node
    int c = idx & 31;                 // float4 chunk (128 = 32 * 4)
    if (i >= n) return;
    float w = dinv[i]; w = w * w;
    float4 v  = ((const float4*)(T + (size_t)i * NF))[c];
    float4 bb = ((const float4*)bias)[c];
    float4 o;
    o.x = bb.x + v.x * w; o.y = bb.y + v.y * w;
    o.z = bb.z + v.z * w; o.w = bb.w + v.w * w;
    ((float4*)(H + (size_t)i * NF))[c] = o;
}

// one wave per edge: coalesced 512B gather of src row, atomic scatter to dst
__global__ void agg_edges_kernel(const float* __restrict__ T,
                                 const float* __restrict__ norm,
                                 const long long* __restrict__ src,
                                 const long long* __restrict__ dst,
                                 float* __restrict__ H, int E) {
    int e = blockIdx.x * (blockDim.x >> 5) + (threadIdx.x >> 5);
    if (e >= E) return;
    int lane = threadIdx.x & 31;
    int s = (int)src[e], d = (int)dst[e];
    float w = norm[e];
    float4 v = ((const float4*)(T + (size_t)s * NF))[lane];
    float* o = H + (size_t)d * NF + lane * 4;
    atomicAdd(o + 0, v.x * w);
    atomicAdd(o + 1, v.y * w);
    atomicAdd(o + 2, v.z * w);
    atomicAdd(o + 3, v.w * w);
}

// ---------------- 128 -> 3 output conv ----------------
__global__ void out_gemm_kernel(const float* __restrict__ H,
                                const float* __restrict__ Wout,
                                float* __restrict__ T3, int n) {
    int i = blockIdx.x * blockDim.x + threadIdx.x;
    if (i >= n) return;
    float a0 = 0.f, a1 = 0.f, a2 = 0.f;
    const float* row = H + (size_t)i * NF;
#pragma unroll 4
    for (int k = 0; k < NF; ++k) {
        float h = row[k];
        h = fmaxf(h, 0.0f) + SLOPE * fminf(h, 0.0f);
        a0 += h * Wout[k * 3 + 0];
        a1 += h * Wout[k * 3 + 1];
        a2 += h * Wout[k * 3 + 2];
    }
    T3[i * 3 + 0] = a0; T3[i * 3 + 1] = a1; T3[i * 3 + 2] = a2;
}

__global__ void out_agg_init_kernel(const float* __restrict__ T3,
                                    const float* __restrict__ dinv,
                                    const float* __restrict__ bout,
                                    float* __restrict__ out, int n) {
    int i = blockIdx.x * blockDim.x + threadIdx.x;
    if (i >= n) return;
    float w = dinv[i]; w = w * w;
    out[i * 3 + 0] = bout[0] + T3[i * 3 + 0] * w;
    out[i * 3 + 1] = bout[1] + T3[i * 3 + 1] * w;
    out[i * 3 + 2] = bout[2] + T3[i * 3 + 2] * w;
}

__global__ void out_agg_edges_kernel(const float* __restrict__ T3,
                                     const float* __restrict__ norm,
                                     const long long* __restrict__ src,
                                     const long long* __restrict__ dst,
                                     float* __restrict__ out, int E) {
    int e = blockIdx.x * blockDim.x + threadIdx.x;
    if (e >= E) return;
    int s = (int)src[e], d = (int)dst[e];
    float w = norm[e];
    atomicAdd(&out[d * 3 + 0], T3[s * 3 + 0] * w);
    atomicAdd(&out[d * 3 + 1], T3[s * 3 + 1] * w);
    atomicAdd(&out[d * 3 + 2], T3[s * 3 + 2] * w);
}

__global__ void tanh_kernel(float* out, int n) {
    int i = blockIdx.x * blockDim.x + threadIdx.x;
    if (i < n) out[i] = tanhf(out[i]) * 0.5f;
}

// ---------------- driver ----------------

extern "C" void kernel_launch(void* const* d_in, const int* in_sizes, int n_in,
                              void* d_out, int out_size, void* d_ws, size_t ws_size,
                              hipStream_t stream) {
    const float*     x    = (const float*)d_in[0];
    const long long* edge = (const long long*)d_in[1];
    const float*     W0   = (const float*)d_in[2];
    const float*     b0   = (const float*)d_in[3];
    const float*     Wh   = (const float*)d_in[4];
    const float*     bh   = (const float*)d_in[5];
    const float*     Wout = (const float*)d_in[6];
    const float*     bout = (const float*)d_in[7];
    float*           out  = (float*)d_out;

    const int IN_CH = 64;
    const int N = in_sizes[0] / IN_CH;     // 30000
    const int E = in_sizes[1] / 2;         // 480000
    const long long* src = edge;           // edge_index[0]
    const long long* dst = edge + E;       // edge_index[1]

    char* ws = (char*)d_ws;
    float* dinv = (float*)(ws + 0);                        // N floats
    float* norm = (float*)(ws + (size_t)(1u << 17));       // E floats
    float* T    = (float*)(ws + (size_t)(1u << 21));       // N*128 floats
    float* H    = (float*)(ws + (size_t)(18u << 20));      // N*128 floats
    float* T3   = T;                                       // N*3 floats (reuse)

    const int TB = 256;
    // normalization (once, reused by all layers)
    deg_init_kernel <<<(N + TB - 1) / TB, TB, 0, stream>>>(dinv, N);
    deg_count_kernel<<<(E + TB - 1) / TB, TB, 0, stream>>>(dst, dinv, E);
    dinv_kernel     <<<(N + TB - 1) / TB, TB, 0, stream>>>(dinv, N);
    edge_norm_kernel<<<(E + TB - 1) / TB, TB, 0, stream>>>(src, dst, dinv, norm, E);

    const int Mtiles    = N / 16;                     // 1875
    int gemm_grid = 256;                              // persistent blocks (8 waves each)
    if (gemm_grid > Mtiles) gemm_grid = Mtiles;
    const int agg_grid  = (N * 32 + TB - 1) / TB;
    const int edge_grid = (E + (TB / 32) - 1) / (TB / 32);

    // layer 0: x[N,64] @ W0 -> aggregate
    gemm_wmma_kernel<64, false><<<gemm_grid, TB, 0, stream>>>(x, W0, T, Mtiles);
    agg_init_kernel <<<agg_grid,  TB, 0, stream>>>(T, dinv, b0, H, N);
    agg_edges_kernel<<<edge_grid, TB, 0, stream>>>(T, norm, src, dst, H, E);

    // 6 hidden layers (leaky-relu fused into GEMM A-load)
    for (int l = 0; l < 6; ++l) {
        gemm_wmma_kernel<128, true><<<gemm_grid, TB, 0, stream>>>(
            H, Wh + (size_t)l * NF * NF, T, Mtiles);
        agg_init_kernel <<<agg_grid,  TB, 0, stream>>>(T, dinv, bh + (size_t)l * NF, H, N);
        agg_edges_kernel<<<edge_grid, TB, 0, stream>>>(T, norm, src, dst, H, E);
    }

    // output conv 128 -> 3, aggregate, tanh * 0.5
    out_gemm_kernel     <<<(N + TB - 1) / TB, TB, 0, stream>>>(H, Wout, T3, N);
    out_agg_init_kernel <<<(N + TB - 1) / TB, TB, 0, stream>>>(T3, dinv, bout, out, N);
    out_agg_edges_kernel<<<(E + TB - 1) / TB, TB, 0, stream>>>(T3, norm, src, dst, out, E);
    tanh_kernel         <<<(N * 3 + TB - 1) / TB, TB, 0, stream>>>(out, N * 3);
}